// SAGE_3985729651444
// MI455X (gfx1250) — compile-verified
//
#include <hip/hip_runtime.h>

typedef float v2f __attribute__((ext_vector_type(2)));
typedef float v8f __attribute__((ext_vector_type(8)));

#define NNODES 50000
#define NEDGES 800000

// ---------------- utility ----------------
__global__ void zero_kernel(float* __restrict__ p, int n) {
  int i = blockIdx.x * blockDim.x + threadIdx.x;
  if (i < n) p[i] = 0.0f;
}

// deg[d] += 1 ; P[d][0:64] += ef[e][0:64]   (layer-invariant, computed once)
__global__ void degP_kernel(const float* __restrict__ ef, const int* __restrict__ dstIdx,
                            float* __restrict__ P, float* __restrict__ deg) {
  int e = blockIdx.x;       // one block (64 threads) per edge
  int f = threadIdx.x;      // 0..63
  int d = dstIdx[e];
  if (f == 0) atomicAdd(deg + d, 1.0f);
  atomicAdd(P + (size_t)d * 64 + f, ef[(size_t)e * 64 + f]);
}

// G[dst[e]][f] += h[src[e]][f]  — gathers/atomics stay resident in 192MB L2
__global__ void scatter_kernel(const float* __restrict__ h, const int* __restrict__ srcIdx,
                               const int* __restrict__ dstIdx, float* __restrict__ G, int F) {
  int e = blockIdx.x;       // one block per edge
  int f = threadIdx.x;
  if (f >= F) return;
  int s = srcIdx[e];
  int d = dstIdx[e];
  atomicAdd(G + (size_t)d * F + f, h[(size_t)s * F + f]);
}

// Y[M,Nout] = epilogue( X1[M,K1] @ W[:, 0:K1].T + X2[M,K2] @ W[:, K1:K1+K2].T )
// W row-major [Nout, ldw], ldw = K1+K2.
// One wave (32 threads) computes a 16x32 slab of C: two 16x16 WMMA tiles
// sharing the A fragment (A loaded once per k-step, 2 independent wmma chains).
// Out-of-range W rows are CLAMPED (not zeroed) — the resulting garbage columns
// are never stored (store guard n < Nout), and clamping keeps the inner loop
// branch-free (no exec-mask save/restore per load).
// mode 0: val = (acc + deg[m]*bias[n]) / max(deg[m],1)   (fused mean + msg bias)
// mode 1: val = relu(acc + bias[n])                      (fused update epilogue)
__global__ void gemm_dual_kernel(const float* __restrict__ X1, int ld1, int K1,
                                 const float* __restrict__ X2, int ld2, int K2,
                                 const float* __restrict__ W, int ldw, int Nout,
                                 const float* __restrict__ bias, const float* __restrict__ deg,
                                 int mode, float* __restrict__ Y, int ldy) {
  const int lane = threadIdx.x;            // blockDim.x == 32 (wave32)
  const int m0 = blockIdx.x * 16;
  const int n0 = blockIdx.y * 32;
  const int r  = lane & 15;                // M row (A) / W row (B)
  const int kh = (lane >> 4) << 1;         // k sub-offset: lanes 0-15 -> {k,k+1}, 16-31 -> {k+2,k+3}

  v8f acc0 = {0.f, 0.f, 0.f, 0.f, 0.f, 0.f, 0.f, 0.f};
  v8f acc1 = {0.f, 0.f, 0.f, 0.f, 0.f, 0.f, 0.f, 0.f};

  const int wrow0 = min(n0 + r,      Nout - 1);   // clamped: branch-free B loads
  const int wrow1 = min(n0 + 16 + r, Nout - 1);
  const float* __restrict__ wr0 = W + (size_t)wrow0 * ldw;
  const float* __restrict__ wr1 = W + (size_t)wrow1 * ldw;
  const float* __restrict__ xr1 = X1 + (size_t)(m0 + r) * ld1;

  for (int k = 0; k < K1; k += 4) {
    v2f a  = *(const v2f*)(xr1 + k + kh);
    v2f b0 = *(const v2f*)(wr0 + k + kh);
    v2f b1 = *(const v2f*)(wr1 + k + kh);
    acc0 = __builtin_amdgcn_wmma_f32_16x16x4_f32(false, a, false, b0, (short)0, acc0, false, false);
    acc1 = __builtin_amdgcn_wmma_f32_16x16x4_f32(false, a, false, b1, (short)0, acc1, false, false);
  }
  if (K2 > 0) {
    const float* __restrict__ xr2 = X2 + (size_t)(m0 + r) * ld2;
    const float* __restrict__ wq0 = wr0 + K1;
    const float* __restrict__ wq1 = wr1 + K1;
    for (int k = 0; k < K2; k += 4) {
      v2f a  = *(const v2f*)(xr2 + k + kh);
      v2f b0 = *(const v2f*)(wq0 + k + kh);
      v2f b1 = *(const v2f*)(wq1 + k + kh);
      acc0 = __builtin_amdgcn_wmma_f32_16x16x4_f32(false, a, false, b0, (short)0, acc0, false, false);
      acc1 = __builtin_amdgcn_wmma_f32_16x16x4_f32(false, a, false, b1, (short)0, acc1, false, false);
    }
  }

  // C layout: N = lane&15 (+ tile offset), M = v + 8*(lane>>4)
  const int mh = (lane >> 4) << 3;
  const int nA = n0 + r;
  const int nB = n0 + 16 + r;
#pragma unroll
  for (int v = 0; v < 8; ++v) {
    const int m = m0 + v + mh;
    float dg = 0.0f, inv = 1.0f;
    if (mode == 0) {
      dg  = deg[m];
      inv = 1.0f / fmaxf(dg, 1.0f);
    }
    if (nA < Nout) {
      float bn  = bias[nA];
      float val = acc0[v];
      val = (mode == 0) ? (val + dg * bn) * inv : fmaxf(val + bn, 0.0f);
      Y[(size_t)m * ldy + nA] = val;
    }
    if (nB < Nout) {
      float bn  = bias[nB];
      float val = acc1[v];
      val = (mode == 0) ? (val + dg * bn) * inv : fmaxf(val + bn, 0.0f);
      Y[(size_t)m * ldy + nB] = val;
    }
  }
}

// ---------------- host ----------------
static inline void run_gemm(const float* X1, int ld1, int K1,
                            const float* X2, int ld2, int K2,
                            const float* W, int ldw, int Nout,
                            const float* bias, const float* deg, int mode,
                            float* Y, int ldy, hipStream_t stream) {
  dim3 grid(NNODES / 16, (Nout + 31) / 32);  // 50000 = 3125*16 exactly
  gemm_dual_kernel<<<grid, 32, 0, stream>>>(X1, ld1, K1, X2, ld2, K2,
                                            W, ldw, Nout, bias, deg, mode, Y, ldy);
}

static inline void zero_buf(float* p, size_t n, hipStream_t stream) {
  zero_kernel<<<(int)((n + 255) / 256), 256, 0, stream>>>(p, (int)n);
}

extern "C" void kernel_launch(void* const* d_in, const int* in_sizes, int n_in,
                              void* d_out, int out_size, void* d_ws, size_t ws_size,
                              hipStream_t stream) {
  (void)in_sizes; (void)n_in; (void)out_size; (void)ws_size;
  const float* nfeats = (const float*)d_in[0];   // [N,64]
  const float* efeats = (const float*)d_in[1];   // [E,64]
  const float* Wm1 = (const float*)d_in[2];  const float* bm1 = (const float*)d_in[3];
  const float* Wa1 = (const float*)d_in[4];  const float* ba1 = (const float*)d_in[5];
  const float* Wm2 = (const float*)d_in[6];  const float* bm2 = (const float*)d_in[7];
  const float* Wa2 = (const float*)d_in[8];  const float* ba2 = (const float*)d_in[9];
  const float* Wm3 = (const float*)d_in[10]; const float* bm3 = (const float*)d_in[11];
  const float* Wa3 = (const float*)d_in[12]; const float* ba3 = (const float*)d_in[13];
  const int*   src = (const int*)d_in[14];
  const int*   dst = (const int*)d_in[15];
  float* out = (float*)d_out;

  // workspace layout
  float* ws  = (float*)d_ws;
  float* G   = ws;                            // [N,152] (ld = F per layer)
  float* P   = G   + (size_t)NNODES * 152;    // [N,64]  edge-feature sums (layer-invariant)
  float* deg = P   + (size_t)NNODES * 64;     // [N]
  float* hn  = deg + (size_t)NNODES;          // [N,152] h_neigh
  float* hA  = hn  + (size_t)NNODES * 152;    // [N,152] layer outputs (ping)
  float* hB  = hA  + (size_t)NNODES * 152;    // [N,152] layer outputs (pong)

  // ---- invariant precompute: deg, P ----
  zero_buf(P, (size_t)NNODES * 64 + NNODES, stream);  // zeros P and deg (contiguous)
  degP_kernel<<<NEDGES, 64, 0, stream>>>(efeats, dst, P, deg);

  // ---- layer 1: Fin=64, HID=152 ----
  zero_buf(G, (size_t)NNODES * 64, stream);
  scatter_kernel<<<NEDGES, 64, 0, stream>>>(nfeats, src, dst, G, 64);
  run_gemm(G, 64, 64,  P, 64, 64,   Wm1, 128, 152, bm1, deg, 0, hn, 152, stream);
  run_gemm(nfeats, 64, 64, hn, 152, 152, Wa1, 216, 152, ba1, nullptr, 1, hA, 152, stream);

  // ---- layer 2: Fin=152, HID=152 ----
  zero_buf(G, (size_t)NNODES * 152, stream);
  scatter_kernel<<<NEDGES, 160, 0, stream>>>(hA, src, dst, G, 152);
  run_gemm(G, 152, 152, P, 64, 64,  Wm2, 216, 152, bm2, deg, 0, hn, 152, stream);
  run_gemm(hA, 152, 152, hn, 152, 152, Wa2, 304, 152, ba2, nullptr, 1, hB, 152, stream);

  // ---- layer 3: Fin=152, OUT=64 ----
  zero_buf(G, (size_t)NNODES * 152, stream);
  scatter_kernel<<<NEDGES, 160, 0, stream>>>(hB, src, dst, G, 152);
  run_gemm(G, 152, 152, P, 64, 64,  Wm3, 216, 64, bm3, deg, 0, hn, 64, stream);
  run_gemm(hB, 152, 152, hn, 64, 64, Wa3, 216, 64, ba3, nullptr, 1, out, 64, stream);
}